// GeomGCNSingleChannel_68049461838274
// MI455X (gfx1250) — compile-verified
//
#include <hip/hip_runtime.h>
#include <hip/hip_bf16.h>

// ---------------------------------------------------------------------------
// GeomGCN single channel for MI455X (gfx1250, wave32, WMMA)
//
//   Wh[d,n,:]   = (feature[n,:] @ W[d,:,:]^T) * norm[n]         (WMMA f32 GEMM)
//   out[dst*576 + d*64 + o] += Wh[d, src, o]   over edges        (atomic scatter)
//   out = relu(out * norm[n])                                    (finalize)
// ---------------------------------------------------------------------------

#define NNODES   100000
#define IN_F     128
#define OUT_F    64
#define NDIV     9
#define MTILES   (NNODES / 16)          // 6250, exact
#define TOTAL_OUT (NNODES * NDIV * OUT_F)

typedef __attribute__((ext_vector_type(2))) float v2f;
typedef __attribute__((ext_vector_type(8))) float v8f;

// -------------------------- 1) WMMA GEMM --------------------------
// One wave computes a 16-node x 64-out tile for one division.
// wave id = mt + MTILES * d   (adjacent waves stream adjacent node tiles).
__global__ __launch_bounds__(256)
void gcn_gemm_wmma(const float* __restrict__ feat,
                   const float* __restrict__ norm,
                   const float* __restrict__ W,
                   float* __restrict__ Wh)
{
    const int wave = blockIdx.x * (blockDim.x >> 5) + (threadIdx.x >> 5);
    if (wave >= MTILES * NDIV) return;          // wave-uniform: EXEC stays full

    const int lane = threadIdx.x & 31;
    const int m    = lane & 15;                 // row-in-tile / col-in-tile
    const int h    = lane >> 4;                 // half-wave selector (K offset +2)

    const int mt = wave % MTILES;
    const int d  = wave / MTILES;

    // A: feat[mt*16 + m][k0 + 2h .. +1]   B: W[d][col + m][k0 + 2h .. +1]
    const float* arow = feat + (size_t)(mt * 16 + m) * IN_F + 2 * h;
    const float* wrow = W    + ((size_t)d * OUT_F + m) * IN_F + 2 * h;

    v8f acc[4] = {v8f{0}, v8f{0}, v8f{0}, v8f{0}};

#pragma unroll
    for (int k0 = 0; k0 < IN_F; k0 += 4) {
        const v2f a = *(const v2f*)(arow + k0);
#pragma unroll
        for (int nt = 0; nt < 4; ++nt) {
            const v2f b = *(const v2f*)(wrow + (size_t)nt * 16 * IN_F + k0);
            acc[nt] = __builtin_amdgcn_wmma_f32_16x16x4_f32(
                false, a, false, b, (short)0, acc[nt], false, false);
        }
    }

    // D layout: VGPR v -> row (v + 8h), col m.  Scale by norm[row] on store.
#pragma unroll
    for (int v = 0; v < 8; ++v) {
        const int grow = mt * 16 + v + 8 * h;
        const float nn = norm[grow];
        float* orow = Wh + ((size_t)d * NNODES + grow) * OUT_F + m;
#pragma unroll
        for (int nt = 0; nt < 4; ++nt)
            orow[nt * 16] = acc[nt][v] * nn;
    }
}

// -------------------------- 2) edge scatter --------------------------
// 16 lanes per edge, float4 per lane (64 outs). Atomic add into d_out with
// the reference's reshape layout: out[dst*576 + div*64 + o].
__global__ __launch_bounds__(256)
void gcn_scatter(const float* __restrict__ Wh,
                 const int* __restrict__ src,
                 const int* __restrict__ dst,
                 const int* __restrict__ ediv,
                 float* __restrict__ out,
                 int nEdges)
{
    const long long t = (long long)blockIdx.x * blockDim.x + threadIdx.x;
    const int e = (int)(t >> 4);
    if (e >= nEdges) return;
    const int q  = ((int)t & 15) * 4;

    const int s  = src[e];
    const int dn = dst[e];
    const int dv = ediv[e];

    const float4 m4 = *(const float4*)(Wh + ((size_t)dv * NNODES + s) * OUT_F + q);
    float* o = out + (size_t)dn * (NDIV * OUT_F) + dv * OUT_F + q;
    atomicAdd(o + 0, m4.x);
    atomicAdd(o + 1, m4.y);
    atomicAdd(o + 2, m4.z);
    atomicAdd(o + 3, m4.w);
}

// -------------------------- 3) finalize --------------------------
// out = relu(out * norm[n]), vectorized float4 (576/4 = 144 chunks per node).
__global__ __launch_bounds__(256)
void gcn_finalize(float* __restrict__ out,
                  const float* __restrict__ norm,
                  int total4)
{
    const int t = blockIdx.x * blockDim.x + threadIdx.x;
    if (t >= total4) return;
    const float nn = norm[t / (NDIV * OUT_F / 4)];   // t / 144
    float4 v = *((float4*)out + t);
    v.x = fmaxf(v.x * nn, 0.0f);
    v.y = fmaxf(v.y * nn, 0.0f);
    v.z = fmaxf(v.z * nn, 0.0f);
    v.w = fmaxf(v.w * nn, 0.0f);
    *((float4*)out + t) = v;
}

// -------------------------- launcher --------------------------
extern "C" void kernel_launch(void* const* d_in, const int* in_sizes, int n_in,
                              void* d_out, int out_size, void* d_ws, size_t ws_size,
                              hipStream_t stream)
{
    const float* feat = (const float*)d_in[0];   // [N, 128]
    const float* norm = (const float*)d_in[1];   // [N, 1]
    const float* W    = (const float*)d_in[2];   // [9, 64, 128]
    const int*   src  = (const int*)d_in[3];     // [E]
    const int*   dst  = (const int*)d_in[4];     // [E]
    const int*   edv  = (const int*)d_in[5];     // [E]
    float*       out  = (float*)d_out;           // [N, 576]
    float*       Wh   = (float*)d_ws;            // [9, N, 64] scratch (230.4 MB)

    const int nEdges = in_sizes[3];

    // 1) Wh = (feature @ W^T) * norm  — 56250 waves, 8 waves / block
    {
        const int waves  = MTILES * NDIV;
        const int blocks = (waves + 7) / 8;
        gcn_gemm_wmma<<<blocks, 256, 0, stream>>>(feat, norm, W, Wh);
    }

    // 2) zero the accumulator (graph-capture-safe stream memset)
    hipMemsetAsync(d_out, 0, (size_t)out_size * sizeof(float), stream);

    // 3) scatter-add messages over edges
    {
        const long long threads = (long long)nEdges * 16;
        const int blocks = (int)((threads + 255) / 256);
        gcn_scatter<<<blocks, 256, 0, stream>>>(Wh, src, dst, edv, out, nEdges);
    }

    // 4) relu(out * norm)
    {
        const int total4 = TOTAL_OUT / 4;
        const int blocks = (total4 + 255) / 256;
        gcn_finalize<<<blocks, 256, 0, stream>>>(out, norm, total4);
    }
}